// GAT_498216206708
// MI455X (gfx1250) — compile-verified
//
#include <hip/hip_runtime.h>
#include <hip/hip_bf16.h>
#include <math.h>
#include <stdint.h>

// ---------------- constants from the reference ----------------
#define GAT_N     20000
#define GAT_IN    512
#define GAT_HID   64
#define GAT_HEADS 8
#define GAT_F1    (GAT_HEADS * GAT_HID)   // 512
#define GAT_NCLS  64
#define NEG_SLOPE 0.2f

typedef float v2f __attribute__((ext_vector_type(2)));
typedef float v8f __attribute__((ext_vector_type(8)));

// ---------------------------------------------------------------------------
// f32 WMMA GEMM: C[M,N] = A[M,K] @ B[K,N]     (K multiple of 4, N/16 tiles)
//
// Block = 256 threads = 8 waves stacked along M, all sharing one 16-wide
// column panel of B.  The panel (K x 16 f32, 32 KB for K=512) is staged into
// LDS once per block with CDNA5 async global->LDS copies
// (GLOBAL_LOAD_ASYNC_TO_LDS_B128, ASYNCcnt, s_wait_asynccnt), then each wave
// runs the K-loop with V_WMMA_F32_16X16X4_F32, A from global (float2 per
// lane), B from LDS (ds_load).
//
// Waves whose M-tile falls past the end clamp their row (wave-uniform scalar
// branch -> EXEC stays all-ones around WMMA) and skip the store.
// ---------------------------------------------------------------------------
#define GEMM_KMAX 512           // both layers have K = 512
__global__ void __launch_bounds__(256)
wmma_gemm_f32(const float* __restrict__ A, const float* __restrict__ B,
              float* __restrict__ C, int M, int N, int K) {
    __shared__ float panelB[GEMM_KMAX * 16];   // [k][0..15], 32 KB

    const int tid    = threadIdx.x;
    const int wave   = tid >> 5;
    const int lane   = tid & 31;
    const int half   = lane >> 4;            // 0: lanes 0-15, 1: lanes 16-31
    const int l      = lane & 15;

    const int Mtiles = M >> 4;
    const int tile_n = blockIdx.y;
    const int tile_m = blockIdx.x * 8 + wave;          // may exceed Mtiles-1
    const int tile_c = (tile_m < Mtiles) ? tile_m : (Mtiles - 1);  // clamp

    const int col0 = tile_n * 16;
    const int row0 = tile_c * 16;

    // ---- stage B panel: K*4 chunks of 16 bytes, 256 threads cooperate ----
    // chunk c: k = c>>2, seg = c&3 -> global B[k*N + col0 + seg*4 .. +3]
    //                                 lds  panelB + c*16 bytes
    const unsigned ldsBase = (unsigned)(uintptr_t)(&panelB[0]);
    const int nChunks = K * 4;
    for (int c = tid; c < nChunks; c += 256) {
        const int k   = c >> 2;
        const int seg = c & 3;
        const uint64_t gaddr =
            (uint64_t)(uintptr_t)(B + (size_t)k * N + col0 + seg * 4);
        const unsigned laddr = ldsBase + (unsigned)c * 16u;
        asm volatile("global_load_async_to_lds_b128 %0, %1, off"
                     :
                     : "v"(laddr), "v"(gaddr)
                     : "memory");
    }
    asm volatile("s_wait_asynccnt 0x0" ::: "memory");
    __syncthreads();

    // ---- per-wave 16x16 tile, K stepped by 4 ----
    v8f acc = {0.f, 0.f, 0.f, 0.f, 0.f, 0.f, 0.f, 0.f};
    const float* arow = A + (size_t)(row0 + l) * K;
#pragma unroll 4
    for (int k = 0; k < K; k += 4) {
        const int kk = k + half * 2;
        // A (16x4): lanes 0-15 -> K={k,k+1}; lanes 16-31 -> K={k+2,k+3}
        const float* ap = arow + kk;
        v2f a;
        a.x = ap[0];
        a.y = ap[1];
        // B (4x16) from LDS: same K split, column col0+l
        v2f b;
        b.x = panelB[kk * 16 + l];
        b.y = panelB[(kk + 1) * 16 + l];
        acc = __builtin_amdgcn_wmma_f32_16x16x4_f32(
            /*neg_a=*/false, a, /*neg_b=*/false, b,
            /*c_mod=*/(short)0, acc, /*reuse_a=*/false, /*reuse_b=*/false);
    }

    // D layout: VGPR j -> row = row0 + half*8 + j, col = col0 + l
    if (tile_m < Mtiles) {                 // wave-uniform
#pragma unroll
        for (int j = 0; j < 8; ++j) {
            C[(size_t)(row0 + half * 8 + j) * N + col0 + l] = acc[j];
        }
    }
}

// ---------------------------------------------------------------------------
// Per-node attention scores: als[n,h] = <h[n,h,:], a_src[h,:]>, same for ald.
// ---------------------------------------------------------------------------
__global__ void scores_kernel(const float* __restrict__ h,
                              const float* __restrict__ a_src,
                              const float* __restrict__ a_dst,
                              float* __restrict__ als, float* __restrict__ ald,
                              int Nn, int Hh, int C) {
    int i = blockIdx.x * blockDim.x + threadIdx.x;
    if (i >= Nn * Hh) return;
    int n  = i / Hh;
    int hd = i - n * Hh;
    const float* hp = h + (size_t)n * Hh * C + (size_t)hd * C;
    const float* as = a_src + (size_t)hd * C;
    const float* ad = a_dst + (size_t)hd * C;
    float s = 0.f, d = 0.f;
    for (int c = 0; c < C; ++c) {
        float v = hp[c];
        s = fmaf(v, as[c], s);
        d = fmaf(v, ad[c], d);
    }
    als[i] = s;
    ald[i] = d;
}

// ---------------------------------------------------------------------------
// Edge pass A: w[e,h] = exp(leakyrelu(als[src]+ald[dst])); denom[dst,h] += w.
// Self-loops appended: edge id e >= Eorig maps to node (e - Eorig).
// ---------------------------------------------------------------------------
__global__ void edge_w_kernel(const int* __restrict__ src_idx,
                              const int* __restrict__ dst_idx,
                              const float* __restrict__ als,
                              const float* __restrict__ ald,
                              float* __restrict__ w, float* __restrict__ denom,
                              int Etot, int Eorig, int Hh) {
    int i = blockIdx.x * blockDim.x + threadIdx.x;
    if (i >= Etot * Hh) return;
    int e  = i / Hh;
    int hd = i - e * Hh;
    int s = (e < Eorig) ? src_idx[e] : (e - Eorig);
    int d = (e < Eorig) ? dst_idx[e] : (e - Eorig);
    float x = als[(size_t)s * Hh + hd] + ald[(size_t)d * Hh + hd];
    x = (x > 0.f) ? x : NEG_SLOPE * x;
    float ww = __expf(x);         // scores are O(1); shift-free softmax is safe
    w[i] = ww;
    atomicAdd(&denom[(size_t)d * Hh + hd], ww);
}

// ---------------------------------------------------------------------------
// Edge pass B: acc[dst, h*C+c] += h[src, h*C+c] * (w[e,h] / denom[dst,h]).
// One wave per edge; lanes stride the F = Hh*C feature dim.
// ---------------------------------------------------------------------------
__global__ void edge_agg_kernel(const int* __restrict__ src_idx,
                                const int* __restrict__ dst_idx,
                                const float* __restrict__ hfeat,
                                const float* __restrict__ w,
                                const float* __restrict__ denom,
                                float* __restrict__ acc,
                                int Etot, int Eorig, int Hh, int C) {
    int wave = (int)((blockIdx.x * blockDim.x + threadIdx.x) >> 5);
    int lane = threadIdx.x & 31;
    if (wave >= Etot) return;
    int e = wave;
    int s = (e < Eorig) ? src_idx[e] : (e - Eorig);
    int d = (e < Eorig) ? dst_idx[e] : (e - Eorig);
    int F = Hh * C;
    const float* hs = hfeat + (size_t)s * F;
    float*       ad = acc   + (size_t)d * F;
    for (int c = lane; c < F; c += 32) {
        int hd = c / C;
        float alpha = w[(size_t)e * Hh + hd] / denom[(size_t)d * Hh + hd];
        atomicAdd(&ad[c], hs[c] * alpha);
    }
}

// out[n,f] = relu(acc[n,f] + bias[f])
__global__ void relu_bias_kernel(const float* __restrict__ acc,
                                 const float* __restrict__ bias,
                                 float* __restrict__ out, int total, int F) {
    int i = blockIdx.x * blockDim.x + threadIdx.x;
    if (i >= total) return;
    int f = i % F;
    float v = acc[i] + bias[f];
    out[i] = v > 0.f ? v : 0.f;
}

// out[n,f] += bias[f]
__global__ void add_bias_kernel(float* __restrict__ out,
                                const float* __restrict__ bias,
                                int total, int F) {
    int i = blockIdx.x * blockDim.x + threadIdx.x;
    if (i >= total) return;
    out[i] += bias[i % F];
}

__global__ void zero_kernel(float* __restrict__ p, size_t n) {
    size_t i = (size_t)blockIdx.x * blockDim.x + threadIdx.x;
    size_t stride = (size_t)gridDim.x * blockDim.x;
    for (; i < n; i += stride) p[i] = 0.f;
}

static inline int cdiv(long long a, long long b) { return (int)((a + b - 1) / b); }

extern "C" void kernel_launch(void* const* d_in, const int* in_sizes, int n_in,
                              void* d_out, int out_size, void* d_ws, size_t ws_size,
                              hipStream_t stream) {
    const float* x      = (const float*)d_in[0];
    const int*   eidx   = (const int*)  d_in[1];   // [2, E] row-major
    const float* W1     = (const float*)d_in[2];
    const float* a_src1 = (const float*)d_in[3];
    const float* a_dst1 = (const float*)d_in[4];
    const float* b1     = (const float*)d_in[5];
    const float* W2     = (const float*)d_in[6];
    const float* a_src2 = (const float*)d_in[7];
    const float* a_dst2 = (const float*)d_in[8];
    const float* b2     = (const float*)d_in[9];
    float* out = (float*)d_out;

    const int N    = GAT_N;
    const int E    = in_sizes[1] / 2;     // 320000
    const int Etot = E + N;               // + self-loops
    const int IN   = GAT_IN;
    const int F1   = GAT_F1;              // 512
    const int NC   = GAT_NCLS;            // 64

    const int* srcp = eidx;               // edge_index[0]
    const int* dstp = eidx + E;           // edge_index[1]

    // ---- workspace carve-up (floats) ----
    float* ws = (float*)d_ws;
    size_t o = 0;
    float* h1     = ws + o; o += (size_t)N * F1;      // layer-1 features (pre-agg), reused as h2
    float* acc1   = ws + o; o += (size_t)N * F1;      // layer-1 aggregation
    float* als1   = ws + o; o += (size_t)N * GAT_HEADS;
    float* ald1   = ws + o; o += (size_t)N * GAT_HEADS;
    float* denom1 = ws + o; o += (size_t)N * GAT_HEADS;
    float* w1     = ws + o; o += (size_t)Etot * GAT_HEADS;
    float* g2     = ws + o; o += (size_t)N * NC;      // layer-2 features (pre-agg)
    float* als2   = ws + o; o += (size_t)N;
    float* ald2   = ws + o; o += (size_t)N;
    float* denom2 = ws + o; o += (size_t)N;
    float* w2     = ws + o; o += (size_t)Etot;
    (void)ws_size; (void)n_in; (void)out_size;

    const int ZB = 4096;  // grid-stride zero fill
    zero_kernel<<<ZB, 256, 0, stream>>>(acc1,   (size_t)N * F1);
    zero_kernel<<<ZB, 256, 0, stream>>>(denom1, (size_t)N * GAT_HEADS);
    zero_kernel<<<ZB, 256, 0, stream>>>(denom2, (size_t)N);
    zero_kernel<<<ZB, 256, 0, stream>>>(out,    (size_t)N * NC);

    const int Mtiles = N / 16;            // 1250

    // ===== Layer 1 =====
    // h1 = x @ W1   (bias is applied after aggregation, per reference)
    {
        dim3 grid(cdiv(Mtiles, 8), F1 / 16);
        wmma_gemm_f32<<<grid, 256, 0, stream>>>(x, W1, h1, N, F1, IN);
    }
    scores_kernel<<<cdiv((long long)N * GAT_HEADS, 256), 256, 0, stream>>>(
        h1, a_src1, a_dst1, als1, ald1, N, GAT_HEADS, GAT_HID);
    edge_w_kernel<<<cdiv((long long)Etot * GAT_HEADS, 256), 256, 0, stream>>>(
        srcp, dstp, als1, ald1, w1, denom1, Etot, E, GAT_HEADS);
    edge_agg_kernel<<<cdiv((long long)Etot * 32, 256), 256, 0, stream>>>(
        srcp, dstp, h1, w1, denom1, acc1, Etot, E, GAT_HEADS, GAT_HID);
    // h2 = relu(acc1 + b1), stored back into the h1 buffer (h1 dead now)
    relu_bias_kernel<<<cdiv((long long)N * F1, 256), 256, 0, stream>>>(
        acc1, b1, h1, N * F1, F1);

    // ===== Layer 2 =====
    // g2 = h2 @ W2
    {
        dim3 grid(cdiv(Mtiles, 8), NC / 16);
        wmma_gemm_f32<<<grid, 256, 0, stream>>>(h1, W2, g2, N, NC, F1);
    }
    scores_kernel<<<cdiv((long long)N, 256), 256, 0, stream>>>(
        g2, a_src2, a_dst2, als2, ald2, N, 1, NC);
    edge_w_kernel<<<cdiv((long long)Etot, 256), 256, 0, stream>>>(
        srcp, dstp, als2, ald2, w2, denom2, Etot, E, 1);
    edge_agg_kernel<<<cdiv((long long)Etot * 32, 256), 256, 0, stream>>>(
        srcp, dstp, g2, w2, denom2, out, Etot, E, 1, NC);
    // heads=1, concat=False -> mean over 1 head is identity; just add bias
    add_bias_kernel<<<cdiv((long long)N * NC, 256), 256, 0, stream>>>(
        out, b2, N * NC, NC);
}